// FlashMHA_49271864819788
// MI455X (gfx1250) — compile-verified
//
#include <hip/hip_runtime.h>
#include <hip/hip_bf16.h>

// MHA forward with RoPE for MI455X (gfx1250), fp32 end-to-end using
// V_WMMA_F32_16X16X4_F32 (full-precision matrix path).
//
// Shapes: C=4, L=2048, D=1024, H=16, HD=64.  Workspace layout (floats):
//   Q   : [8192,1024] @ ws + 0
//   K   : [8192,1024] @ ws + 8M
//   V   : [8192,1024] @ ws + 16M
//   ctx : [8192,1024] @ ws + 24M     (total 134.2 MB)

typedef __attribute__((ext_vector_type(2))) float v2f;
typedef __attribute__((ext_vector_type(8))) float v8f;

#define WMMA_F32(a, b, c) \
  __builtin_amdgcn_wmma_f32_16x16x4_f32(false, (a), false, (b), (short)0, (c), false, false)

// ---------------------------------------------------------------------------
// Generic Out[M,N] = A[M,K] @ W[N,K]^T + bias[N]
// Block: 256 threads (8 waves), tile 128x64; each wave owns 32x32 (2x2 WMMA).
// LDS rows padded to 20 floats (80B: 16B-aligned, conflict-free frag reads).
// ---------------------------------------------------------------------------
__global__ __launch_bounds__(256) void gemm_bias_wmma(
    const float* __restrict__ A, const float* __restrict__ W,
    const float* __restrict__ bias, float* __restrict__ Out,
    int M, int N, int K)
{
  __shared__ __align__(16) float As[128 * 20];
  __shared__ __align__(16) float Ws[64 * 20];

  const int tid  = threadIdx.x;
  const int lane = tid & 31, lo = lane & 15, hi = lane >> 4;
  const int wave = tid >> 5, wm = wave & 3, wn = wave >> 2;
  const int nbn  = N >> 6;
  const int bm   = blockIdx.x / nbn;
  const int bn   = blockIdx.x % nbn;

  v8f acc[2][2] = {{{}, {}}, {{}, {}}};

  for (int kb = 0; kb < K; kb += 16) {
    __syncthreads();
    // stage A tile: 128x16 floats, 2 float4 per thread
#pragma unroll
    for (int r = 0; r < 2; ++r) {
      int idx = tid + 256 * r;
      int row = idx >> 2, c4 = idx & 3;
      *(float4*)&As[row * 20 + c4 * 4] =
          *(const float4*)&A[(size_t)(bm * 128 + row) * K + kb + c4 * 4];
    }
    // stage W tile: 64x16 floats, 1 float4 per thread
    {
      int row = tid >> 2, c4 = tid & 3;
      *(float4*)&Ws[row * 20 + c4 * 4] =
          *(const float4*)&W[(size_t)(bn * 64 + row) * K + kb + c4 * 4];
    }
    __syncthreads();

#pragma unroll
    for (int kc = 0; kc < 4; ++kc) {
      const int k0 = kc * 4 + 2 * hi;  // A/B frag: K = 2*hi + elem (ISA 16x4 layout)
      v2f a0 = *(const v2f*)&As[(wm * 32 +  0 + lo) * 20 + k0];
      v2f a1 = *(const v2f*)&As[(wm * 32 + 16 + lo) * 20 + k0];
      v2f b0 = *(const v2f*)&Ws[(wn * 32 +  0 + lo) * 20 + k0];
      v2f b1 = *(const v2f*)&Ws[(wn * 32 + 16 + lo) * 20 + k0];
      acc[0][0] = WMMA_F32(a0, b0, acc[0][0]);
      acc[0][1] = WMMA_F32(a0, b1, acc[0][1]);
      acc[1][0] = WMMA_F32(a1, b0, acc[1][0]);
      acc[1][1] = WMMA_F32(a1, b1, acc[1][1]);
    }
  }

  // C/D layout: VGPR r holds row (r + 8*hi), column = lo
#pragma unroll
  for (int i = 0; i < 2; ++i)
#pragma unroll
    for (int j = 0; j < 2; ++j) {
      int n = bn * 64 + wn * 32 + j * 16 + lo;
      float bv = bias[n];
#pragma unroll
      for (int r = 0; r < 8; ++r) {
        int m = bm * 128 + wm * 32 + i * 16 + r + 8 * hi;
        Out[(size_t)m * N + n] = acc[i][j][r] + bv;
      }
    }
}

// ---------------------------------------------------------------------------
// RoPE applied in-place to Q and K. One thread per (c,l,h,pair).
// q[c,l,h,2i], q[c,l,h,2i+1] rotated by angle l * 10000^(-2i/HD).
// ---------------------------------------------------------------------------
__global__ __launch_bounds__(256) void rope_kernel(float* __restrict__ Q,
                                                   float* __restrict__ K)
{
  int idx = blockIdx.x * 256 + threadIdx.x;  // exactly 2^22 threads launched
  int i = idx & 31;
  int h = (idx >> 5) & 15;
  int l = (idx >> 9) & 2047;
  int c = idx >> 20;
  float inv = __expf(-((float)(2 * i) * (1.0f / 64.0f)) * 9.210340371976184f);
  float ang = (float)l * inv;
  float sn, cs;
  __sincosf(ang, &sn, &cs);
  size_t off = ((size_t)(c * 2048 + l)) * 1024 + h * 64 + 2 * i;
  float2 q = *(float2*)&Q[off];
  float2 k = *(float2*)&K[off];
  *(float2*)&Q[off] = make_float2(q.x * cs - q.y * sn, q.x * sn + q.y * cs);
  *(float2*)&K[off] = make_float2(k.x * cs - k.y * sn, k.x * sn + k.y * cs);
}

// ---------------------------------------------------------------------------
// Flash attention per (c,h): ctx = softmax(mask(Q K^T * 0.125)) @ V.
// Block: 128 threads (4 waves); block covers 64 q-rows, each wave 16 rows.
// K/V 16x64 tiles staged in LDS (row pad 68 = 272B, 16B aligned).
// P (16x16, C-layout in regs) goes through per-wave LDS (row pad 18) to be
// re-read in A-fragment layout for the PV WMMAs.
// ---------------------------------------------------------------------------
__global__ __launch_bounds__(128) void flash_attn_wmma(
    const float* __restrict__ Q, const float* __restrict__ K,
    const float* __restrict__ V, const unsigned char* __restrict__ mask,
    float* __restrict__ ctx)
{
  __shared__ __align__(16) float Ks[16 * 68];
  __shared__ __align__(16) float Vs[16 * 68];
  __shared__ __align__(16) float Ps[4 * 16 * 18];

  const int tid  = threadIdx.x;
  const int lane = tid & 31, lo = lane & 15, hi = lane >> 4;
  const int wv   = tid >> 5;
  const int bx   = blockIdx.x;
  const int qb   = bx & 31;         // 32 q-tiles of 64 rows
  const int h    = (bx >> 5) & 15;
  const int c    = bx >> 9;
  const int cL   = c * 2048;
  const int q0   = qb * 64;
  const size_t headoff = (size_t)h * 64;

  // Q as A-fragments for all 16 K-chunks of the d=64 reduction.
  // A 16x4 layout: lane row = lo, K = 4t + 2*hi + elem.
  v2f qA[16];
  {
    const size_t qrow = (size_t)(cL + q0 + wv * 16 + lo) * 1024 + headoff;
#pragma unroll
    for (int t = 0; t < 16; ++t)
      qA[t] = *(const v2f*)&Q[qrow + 4 * t + 2 * hi];
  }

  v8f o0 = {}, o1 = {}, o2 = {}, o3 = {};
  float mrun[8], lrun[8];
#pragma unroll
  for (int r = 0; r < 8; ++r) { mrun[r] = -1e30f; lrun[r] = 0.0f; }

  float* Pw = &Ps[wv * 16 * 18];

  for (int kb = 0; kb < 2048; kb += 16) {
    __syncthreads();
    // stage K/V tile (16 keys x 64 dims), 2 float4 per thread per buffer
#pragma unroll
    for (int r = 0; r < 2; ++r) {
      int idx = tid + 128 * r;
      int row = idx >> 4, c4 = idx & 15;
      size_t g = (size_t)(cL + kb + row) * 1024 + headoff + c4 * 4;
      *(float4*)&Ks[row * 68 + c4 * 4] = *(const float4*)&K[g];
      *(float4*)&Vs[row * 68 + c4 * 4] = *(const float4*)&V[g];
    }
    __syncthreads();

    // S(16x16) = Q(16x64) @ K^T(64x16): 16 fp32 WMMAs
    v8f s = {};
#pragma unroll
    for (int t = 0; t < 16; ++t) {
      v2f b = *(const v2f*)&Ks[lo * 68 + 4 * t + 2 * hi];  // B: n=lo, k=4t+2hi+elem
      s = WMMA_F32(qA[t], b, s);
    }

    // key-padding mask: this lane's column is key (kb + lo)
    const bool masked = (mask[cL + kb + lo] == 0);

    // online softmax; lane owns rows (r + 8*hi), its 16 row-elements live
    // across the 16-lane half -> shfl_xor 1/2/4/8 reductions stay in-half.
#pragma unroll
    for (int r = 0; r < 8; ++r) {
      float v = masked ? -1e30f : s[r] * 0.125f;
      float rm = v;
      rm = fmaxf(rm, __shfl_xor(rm, 1));
      rm = fmaxf(rm, __shfl_xor(rm, 2));
      rm = fmaxf(rm, __shfl_xor(rm, 4));
      rm = fmaxf(rm, __shfl_xor(rm, 8));
      float mnew = fmaxf(mrun[r], rm);
      float p = __expf(v - mnew);
      float rs = p;
      rs += __shfl_xor(rs, 1);
      rs += __shfl_xor(rs, 2);
      rs += __shfl_xor(rs, 4);
      rs += __shfl_xor(rs, 8);
      float alpha = __expf(mrun[r] - mnew);
      lrun[r] = lrun[r] * alpha + rs;
      mrun[r] = mnew;
      o0[r] *= alpha; o1[r] *= alpha; o2[r] *= alpha; o3[r] *= alpha;
      Pw[(r + 8 * hi) * 18 + lo] = p;  // C-layout -> LDS for A-layout reread
    }
    __syncthreads();

    // O(16x64) += P(16x16) @ V(16x64): 16 fp32 WMMAs
#pragma unroll
    for (int t = 0; t < 4; ++t) {
      v2f a = *(const v2f*)&Pw[lo * 18 + 4 * t + 2 * hi];
      int r0 = (4 * t + 2 * hi) * 68, r1 = r0 + 68;
      v2f b0, b1, b2, b3;
      b0[0] = Vs[r0 +  0 + lo]; b0[1] = Vs[r1 +  0 + lo];
      b1[0] = Vs[r0 + 16 + lo]; b1[1] = Vs[r1 + 16 + lo];
      b2[0] = Vs[r0 + 32 + lo]; b2[1] = Vs[r1 + 32 + lo];
      b3[0] = Vs[r0 + 48 + lo]; b3[1] = Vs[r1 + 48 + lo];
      o0 = WMMA_F32(a, b0, o0);
      o1 = WMMA_F32(a, b1, o1);
      o2 = WMMA_F32(a, b2, o2);
      o3 = WMMA_F32(a, b3, o3);
    }
  }

  // normalize and store; C-layout rows (r + 8*hi), columns 16*j2 + lo
#pragma unroll
  for (int r = 0; r < 8; ++r) {
    float inv = 1.0f / lrun[r];
    size_t row = (size_t)(cL + q0 + wv * 16 + r + 8 * hi) * 1024 + headoff + lo;
    ctx[row +  0] = o0[r] * inv;
    ctx[row + 16] = o1[r] * inv;
    ctx[row + 32] = o2[r] * inv;
    ctx[row + 48] = o3[r] * inv;
  }
}

// ---------------------------------------------------------------------------
extern "C" void kernel_launch(void* const* d_in, const int* in_sizes, int n_in,
                              void* d_out, int out_size, void* d_ws, size_t ws_size,
                              hipStream_t stream)
{
  const float*         x    = (const float*)d_in[0];
  const unsigned char* mask = (const unsigned char*)d_in[1];  // jnp bool = 1 byte
  const float*         Wq   = (const float*)d_in[2];
  const float*         bq   = (const float*)d_in[3];
  const float*         Wk   = (const float*)d_in[4];
  const float*         bk   = (const float*)d_in[5];
  const float*         Wv   = (const float*)d_in[6];
  const float*         bv   = (const float*)d_in[7];
  const float*         Wo   = (const float*)d_in[8];
  const float*         bo   = (const float*)d_in[9];
  float*               out  = (float*)d_out;

  const int M = 8192, N = 1024, Kd = 1024;
  const size_t sz = (size_t)M * N;
  float* Qb  = (float*)d_ws;
  float* Kb  = Qb + sz;
  float* Vb  = Kb + sz;
  float* ctx = Vb + sz;

  const dim3 gemmGrid((M / 128) * (N / 64));   // 1024 blocks
  gemm_bias_wmma<<<gemmGrid, 256, 0, stream>>>(x, Wq, bq, Qb, M, N, Kd);
  gemm_bias_wmma<<<gemmGrid, 256, 0, stream>>>(x, Wk, bk, Kb, M, N, Kd);
  gemm_bias_wmma<<<gemmGrid, 256, 0, stream>>>(x, Wv, bv, Vb, M, N, Kd);

  rope_kernel<<<16384, 256, 0, stream>>>(Qb, Kb);  // 4*2048*16*32 pairs

  flash_attn_wmma<<<2048, 128, 0, stream>>>(Qb, Kb, Vb, mask, ctx);  // C*H*(L/64)

  gemm_bias_wmma<<<gemmGrid, 256, 0, stream>>>(ctx, Wo, bo, out, M, N, Kd);
}